// OneHotEncoding_22625887715452
// MI455X (gfx1250) — compile-verified
//
#include <hip/hip_runtime.h>

typedef __attribute__((ext_vector_type(2))) float v2f;
typedef __attribute__((ext_vector_type(8))) float v8f;

#define WAVES_PER_BLOCK 8
#define CHUNK_BLOCKS    64          // blocks along the point dimension
#define BIGF            3.0e38f

// ---------------------------------------------------------------------------
// Phase 1: per-(block, receiver-group) partial argmin via V_WMMA_F32_16X16X4_F32
// Score(l, b) = x_l*(-2 rx_b) + y_l*(-2 ry_b) + (x_l^2 + y_l^2) * 1
//             = d2(l,b) - (rx_b^2 + ry_b^2)   -> same argmin as d2, exact f32.
// ---------------------------------------------------------------------------
__global__ __launch_bounds__(256) void argmin_wmma_kernel(
    const float* __restrict__ mesh,       // (L,2)
    const float* __restrict__ recv,       // (B,3)
    float* __restrict__ ws_val,           // (CHUNK_BLOCKS, Btot)
    int*   __restrict__ ws_idx,           // (CHUNK_BLOCKS, Btot)
    int L, int B, int Btot)
{
    const int lane    = threadIdx.x & 31;
    const int wave    = threadIdx.x >> 5;
    const int lane15  = lane & 15;
    const int hiHalf  = lane >> 4;        // 0: lanes 0-15, 1: lanes 16-31
    const int group   = blockIdx.y;       // 16 receivers per group

    // --- B operand (wave-invariant): lanes 0-15 -> (-2rx, -2ry) [K=0,1],
    //     lanes 16-31 -> (1, 0) [K=2,3]
    v2f bop;
    if (hiHalf == 0) {
        int rb = group * 16 + lane15;
        if (rb > B - 1) rb = B - 1;       // clamp (B==256 in practice)
        bop[0] = -2.0f * recv[3 * rb + 0];
        bop[1] = -2.0f * recv[3 * rb + 1];
    } else {
        bop[0] = 1.0f;
        bop[1] = 0.0f;
    }

    float bestVal = BIGF;
    int   bestIdx = 0;

    const int numTiles   = (L + 15) >> 4;
    const int waveStride = gridDim.x * WAVES_PER_BLOCK;

    for (int t = blockIdx.x * WAVES_PER_BLOCK + wave; t < numTiles; t += waveStride) {
        const int base = t << 4;
        int p = base + lane15;
        const bool valid = (p < L);
        if (p > L - 1) p = L - 1;          // clamped, branch-free load
        const float x = mesh[2 * p + 0];
        const float y = mesh[2 * p + 1];

        // --- A operand: lanes 0-15 -> (x, y) [K=0,1],
        //     lanes 16-31 -> (x^2+y^2 or sentinel, 0) [K=2,3]
        v2f aop;
        if (hiHalf == 0) {
            aop[0] = x;
            aop[1] = y;
        } else {
            aop[0] = valid ? (x * x + y * y) : BIGF;  // pad rows lose argmin
            aop[1] = 0.0f;
        }

        v8f c = {};
        c = __builtin_amdgcn_wmma_f32_16x16x4_f32(
                /*neg_a=*/false, aop, /*neg_b=*/false, bop,
                /*c_mod=*/(short)0, c, /*reuse_a=*/false, /*reuse_b=*/false);

        // D layout: lane n (n<16) holds rows 0..7 of column n in c[0..7];
        //           lane n+16 holds rows 8..15 of column n.
        #pragma unroll
        for (int i = 0; i < 8; ++i) {
            const float v   = c[i];
            const int   idx = base + (hiHalf ? (8 + i) : i);
            if (v < bestVal) { bestVal = v; bestIdx = idx; }  // ascending idx per lane
        }
    }

    // --- merge lane n with lane n+16 (same receiver), prefer lower index on tie
    {
        const float oVal = __shfl_xor(bestVal, 16, 32);
        const int   oIdx = __shfl_xor(bestIdx, 16, 32);
        if (oVal < bestVal || (oVal == bestVal && oIdx < bestIdx)) {
            bestVal = oVal; bestIdx = oIdx;
        }
    }

    // --- merge the 8 waves of this block through LDS
    __shared__ float s_val[WAVES_PER_BLOCK][16];
    __shared__ int   s_idx[WAVES_PER_BLOCK][16];
    if (lane < 16) {
        s_val[wave][lane15] = bestVal;
        s_idx[wave][lane15] = bestIdx;
    }
    __syncthreads();

    if (threadIdx.x < 16) {
        float bv = s_val[0][threadIdx.x];
        int   bi = s_idx[0][threadIdx.x];
        #pragma unroll
        for (int w = 1; w < WAVES_PER_BLOCK; ++w) {
            const float v = s_val[w][threadIdx.x];
            const int   i2 = s_idx[w][threadIdx.x];
            if (v < bv || (v == bv && i2 < bi)) { bv = v; bi = i2; }
        }
        const int rb = group * 16 + threadIdx.x;
        ws_val[blockIdx.x * Btot + rb] = bv;
        ws_idx[blockIdx.x * Btot + rb] = bi;
    }
}

// ---------------------------------------------------------------------------
// Phase 2: fill input_tensor = [X, Y, 0]  (one-hot scatter happens in phase 3)
// ---------------------------------------------------------------------------
__global__ __launch_bounds__(256) void fill_kernel(
    const float* __restrict__ mesh, float* __restrict__ out, int L)
{
    const int stride = gridDim.x * blockDim.x;
    for (int i = blockIdx.x * blockDim.x + threadIdx.x; i < L; i += stride) {
        const float x = mesh[2 * i + 0];
        const float y = mesh[2 * i + 1];
        out[3 * i + 0] = x;
        out[3 * i + 1] = y;
        out[3 * i + 2] = 0.0f;
    }
}

// ---------------------------------------------------------------------------
// Phase 3: final reduce over block partials; write min_index (int32 bits),
// closest_points, and scatter the one-hot (stream order guarantees fill ran).
// ---------------------------------------------------------------------------
__global__ void final_kernel(
    const float* __restrict__ mesh,
    const float* __restrict__ ws_val,
    const int*   __restrict__ ws_idx,
    float* __restrict__ out, int L, int B, int Btot, int chunkBlocks)
{
    const int rb = threadIdx.x;          // one thread per receiver
    if (rb >= B) return;

    float bv = BIGF;
    int   bi = 0;
    for (int c = 0; c < chunkBlocks; ++c) {
        const float v  = ws_val[c * Btot + rb];
        const int   i2 = ws_idx[c * Btot + rb];
        if (v < bv || (v == bv && i2 < bi)) { bv = v; bi = i2; }
    }

    float* cp = out + (size_t)3 * L;                 // closest_points (B,2)
    int*   mi = (int*)(out + (size_t)3 * L + 2 * B); // min_index (B,) int32
    cp[2 * rb + 0] = mesh[2 * bi + 0];
    cp[2 * rb + 1] = mesh[2 * bi + 1];
    mi[rb] = bi;

    out[(size_t)3 * bi + 2] = 1.0f;      // one_hot[min_index] = 1 (dup-safe)
    if (rb == 0) out[2] = 1.0f;          // one_hot[0] = 1 since B > 1
}

extern "C" void kernel_launch(void* const* d_in, const int* in_sizes, int n_in,
                              void* d_out, int out_size, void* d_ws, size_t ws_size,
                              hipStream_t stream) {
    const float* mesh = (const float*)d_in[0];   // (L,2) f32
    const float* recv = (const float*)d_in[1];   // (B,3) f32
    float* out = (float*)d_out;

    const int L = in_sizes[0] / 2;               // 1,000,000
    const int B = in_sizes[1] / 3;               // 256
    const int numGroups = (B + 15) / 16;         // 16
    const int Btot = numGroups * 16;             // 256

    float* ws_val = (float*)d_ws;
    int*   ws_idx = (int*)((char*)d_ws + (size_t)CHUNK_BLOCKS * Btot * sizeof(float));

    dim3 grid1(CHUNK_BLOCKS, numGroups);
    argmin_wmma_kernel<<<grid1, 256, 0, stream>>>(mesh, recv, ws_val, ws_idx, L, B, Btot);

    fill_kernel<<<512, 256, 0, stream>>>(mesh, out, L);

    final_kernel<<<1, Btot, 0, stream>>>(mesh, ws_val, ws_idx, out, L, B, Btot, CHUNK_BLOCKS);
}